// SphericalExpansion_58669253264172
// MI455X (gfx1250) — compile-verified
//
#include <hip/hip_runtime.h>
#include <hip/hip_bf16.h>
#include <math.h>

// ---- problem constants (match reference) ----
#define NLMAX     3
#define NBASIS    8                       // N_MAX
#define NSPEC     4
#define ATOMS_PER 100
#define PAIRS_PER 2000
#define NM        16                      // (L_MAX+1)^2
#define FEAT      (NSPEC * NBASIS)        // 32
#define ACC_FLOATS (ATOMS_PER * NM * FEAT) // 51200 floats = 200 KiB
#define WAVES     8
#define PAIRS_PER_WAVE (PAIRS_PER / WAVES) // 250
#define CHUNKS    ((PAIRS_PER_WAVE + 31) / 32) // 8

typedef __attribute__((ext_vector_type(2))) float v2f;
typedef __attribute__((ext_vector_type(8))) float v8f;

__global__ __launch_bounds__(256, 1)
void SphericalExpansion_58669253264172_kernel(const int* __restrict__ species,
                                              const int* __restrict__ pairs,
                                              const float* __restrict__ dirv,
                                              float* __restrict__ out)
{
    // LDS: per-structure accumulator + per-wave WMMA staging
    __shared__ float accum[ACC_FLOATS];                 // 204800 B
    __shared__ float sY[WAVES][32][NM];                 // 16384 B
    __shared__ float sR[WAVES][32][NBASIS];             // 8192 B
    __shared__ int   sMeta[WAVES][32];                  // 1024 B

    const int s    = blockIdx.x;
    const int tid  = threadIdx.x;
    const int wave = tid >> 5;
    const int lane = tid & 31;

    // zero the accumulator
    {
        float4* a4 = (float4*)accum;
        const float4 z4 = make_float4(0.f, 0.f, 0.f, 0.f);
        for (int i = tid; i < ACC_FLOATS / 4; i += 256) a4[i] = z4;
    }
    __syncthreads();

    const int pairBase = s * PAIRS_PER + wave * PAIRS_PER_WAVE;

    for (int chunk = 0; chunk < CHUNKS; ++chunk) {
        const int li     = chunk * 32 + lane;
        const bool active = (li < PAIRS_PER_WAVE);

        float Y[NM];
        float Rv[NBASIS];
        int   cc = 0, sp = 0;
        #pragma unroll
        for (int m = 0; m < NM; ++m) Y[m] = 0.f;
        #pragma unroll
        for (int n = 0; n < NBASIS; ++n) Rv[n] = 0.f;

        if (active) {
            const int p = pairBase + li;
            // pair indices (local to structure), neighbor species
            const int2 pr = ((const int2*)pairs)[p];
            cc = pr.x;
            sp = species[s * ATOMS_PER + pr.y];

            const float dx = dirv[3 * p + 0];
            const float dy = dirv[3 * p + 1];
            const float dz = dirv[3 * p + 2];
            const float q  = dx * dx + dy * dy + dz * dz + 1e-12f;
            const float r  = sqrtf(q);
            const float ri = 1.0f / r;
            const float x = dx * ri, y = dy * ri, z = dz * ri;

            // (x+iy)^m real/imag parts
            const float cs2 = x * x - y * y;
            const float sn2 = 2.0f * x * y;
            const float cs3 = cs2 * x - sn2 * y;
            const float sn3 = cs2 * y + sn2 * x;
            const float z2  = z * z;

            // real fully-normalized Y, sphericart ordering (m = -l..l per l)
            Y[0]  = 0.28209479177387814f;
            Y[1]  = 0.4886025119029199f * y;
            Y[2]  = 0.4886025119029199f * z;
            Y[3]  = 0.4886025119029199f * x;
            Y[4]  = 0.5462742152960396f * sn2;
            Y[5]  = 1.0925484305920792f * y * z;
            Y[6]  = 0.31539156525252005f * (3.0f * z2 - 1.0f);
            Y[7]  = 1.0925484305920792f * x * z;
            Y[8]  = 0.5462742152960396f * cs2;
            Y[9]  = 0.5900435899266435f * sn3;
            Y[10] = 1.445305721320277f  * z * sn2;
            Y[11] = 0.4570457994644658f * (5.0f * z2 - 1.0f) * y;
            Y[12] = 0.3731763325901154f * (5.0f * z2 * z - 3.0f * z);
            Y[13] = 0.4570457994644658f * (5.0f * z2 - 1.0f) * x;
            Y[14] = 1.445305721320277f  * z * cs2;
            Y[15] = 0.5900435899266435f * cs3;

            // radial basis: sin(n*pi*r/3) * fcut, n = 1..8 via angle recurrence
            const float t  = 1.0471975511965976f * r;          // pi*r/3
            const float s1 = sinf(t), c1 = cosf(t);
            const float tm = 1.0471975511965976f * fminf(r, 3.0f);
            const float fc = 0.5f * (cosf(tm) + 1.0f);
            float sk = s1, ck = c1;
            Rv[0] = s1 * fc;
            #pragma unroll
            for (int n = 1; n < NBASIS; ++n) {
                const float sn = sk * c1 + ck * s1;
                const float cn = ck * c1 - sk * s1;
                sk = sn; ck = cn;
                Rv[n] = sn * fc;
            }
        }

        // stage this wave's 32 pairs for the matrix pipe
        #pragma unroll
        for (int m = 0; m < NM; ++m) sY[wave][lane][m] = Y[m];
        #pragma unroll
        for (int n = 0; n < NBASIS; ++n) sR[wave][lane][n] = Rv[n];
        sMeta[wave][lane] = (cc << 2) | sp;
        __syncthreads();

        // 16 WMMAs: each computes 2 pairs' 16x8 outer products.
        //   pair p0 -> K=0, occupies N=0..7 ; pair p1 -> K=1, occupies N=8..15
        // A (16x4 f32, 2 VGPRs): vgpr0 = K0 (lanes0-15 M=lane), vgpr1 = K1
        // B (4x16 f32, 2 VGPRs): vgpr0 = row K0, vgpr1 = row K1
        const int N    = lane & 15;
        const int mofs = (lane >> 4) * 8;
        #pragma unroll
        for (int j = 0; j < 16; ++j) {
            const int p0 = 2 * j, p1 = 2 * j + 1;
            v2f a, b;
            a.x = (lane < 16) ? sY[wave][p0][lane] : 0.f;            // K=0 (K=2 zero)
            a.y = (lane < 16) ? sY[wave][p1][lane] : 0.f;            // K=1 (K=3 zero)
            b.x = (lane < 8)  ? sR[wave][p0][lane] : 0.f;            // row K=0: N=0..7
            b.y = (lane >= 8 && lane < 16) ? sR[wave][p1][lane - 8] : 0.f; // row K=1: N=8..15
            v8f c0 = {0.f, 0.f, 0.f, 0.f, 0.f, 0.f, 0.f, 0.f};
            v8f d = __builtin_amdgcn_wmma_f32_16x16x4_f32(
                false, a, false, b, (short)0, c0, false, false);

            // scatter: lane holds D[M = mofs+i, N] for i=0..7
            const int psel = (N < 8) ? p0 : p1;
            const int meta = sMeta[wave][psel];
            const int c    = meta >> 2;
            const int ss   = meta & 3;
            const int n    = N & 7;
            const int base = c * (NM * FEAT) + ss * NBASIS + n;
            #pragma unroll
            for (int i = 0; i < 8; ++i)
                atomicAdd(&accum[base + (mofs + i) * FEAT], d[i]);
        }
        __syncthreads();
    }

    // stream the finished 200 KiB slab to HBM (coalesced float4 stores)
    {
        float4* o4 = (float4*)(out + (size_t)s * ACC_FLOATS);
        const float4* a4 = (const float4*)accum;
        for (int i = tid; i < ACC_FLOATS / 4; i += 256) o4[i] = a4[i];
    }
}

extern "C" void kernel_launch(void* const* d_in, const int* in_sizes, int n_in,
                              void* d_out, int out_size, void* d_ws, size_t ws_size,
                              hipStream_t stream) {
    const int*   species = (const int*)d_in[0];
    const int*   pairs   = (const int*)d_in[3];
    const float* dirv    = (const float*)d_in[6];
    float*       out     = (float*)d_out;
    const int n_struct = in_sizes[3] / (2 * PAIRS_PER);
    SphericalExpansion_58669253264172_kernel<<<n_struct, 256, 0, stream>>>(
        species, pairs, dirv, out);
}